// phasor2_34205119545599
// MI455X (gfx1250) — compile-verified
//
#include <hip/hip_runtime.h>
#include <math.h>

typedef __attribute__((ext_vector_type(2))) float v2f;
typedef __attribute__((ext_vector_type(8))) float v8f;

#define HW 16384            // 128*128 pixel columns
#define MDIM 128            // CROP
#define NFFT 256
#define LDS_STRIDE 33       // 32 lines + 1 pad -> conflict-free banks
#define SPEC_SCALE (1.0f / 16777216.0f)   // 1/256^3 folded into spectral multiply

// ---------------------------------------------------------------------------
// Kernel 1: temporal convolution with cos/sin wavelets (already reversed by
// setup, so plain cross-correlation), with implicit zero padding at tbe.
// feat (2,96,16384) -> data2 (c=2, d=2, 128, 16384)
// ---------------------------------------------------------------------------
__global__ void conv_wave_kernel(const float* __restrict__ feat,
                                 const float* __restrict__ wave,
                                 const int* __restrict__ tbes,
                                 float* __restrict__ data2) {
  const int p = blockIdx.x * blockDim.x + threadIdx.x;  // pixel column
  const int m = blockIdx.y;                             // output time bin
  const int d = blockIdx.z;                             // D index
  const int tbe = tbes[0];
  const float* fd = feat + (long)d * 96 * HW;
  float sc = 0.f, ss = 0.f;
  for (int j = 0; j < 31; ++j) {
    const int t = m + j - 15 - tbe;
    if (t >= 0 && t < 96) {
      const float x = fd[(long)t * HW + p];
      sc = fmaf(wave[j], x, sc);
      ss = fmaf(wave[31 + j], x, ss);
    }
  }
  const long base = (long)m * HW + p;
  data2[((long)(0 * 2 + d)) * MDIM * HW + base] = sc;
  data2[((long)(1 * 2 + d)) * MDIM * HW + base] = ss;
}

// ---------------------------------------------------------------------------
// Kernel 2: batched GEMM  Y[b] = A(128x128) * X[b](128 x 16384) via
// V_WMMA_F32_16X16X4_F32.  One 16x16 D-tile per wave, K-loop of 32 x k4.
// A fragment: lane<16 rows, VGPR pair = (K, K+1); hi lanes carry K+2,K+3.
// ---------------------------------------------------------------------------
__global__ void gemm128_wmma_kernel(const float* __restrict__ A,
                                    const float* __restrict__ X,
                                    float* __restrict__ Y) {
  const int lane = threadIdx.x & 31;
  const int wave = threadIdx.x >> 5;
  const int col  = lane & 15;
  const int hi   = lane >> 4;                      // 0 or 1: selects K pair / M half
  const int n0   = (blockIdx.x * 4 + wave) * 16;   // column tile
  const int m0   = blockIdx.y * 16;                // row tile
  const long boff = (long)blockIdx.z * MDIM * HW;
  const float* __restrict__ Xb = X + boff;
  float* __restrict__ Yb = Y + boff;

  v8f acc = {0.f, 0.f, 0.f, 0.f, 0.f, 0.f, 0.f, 0.f};
  for (int k = 0; k < 128; k += 4) {
    const int ka = k + 2 * hi;
    v2f a, b;
    a.x = A[(m0 + col) * 128 + ka];
    a.y = A[(m0 + col) * 128 + ka + 1];
    b.x = Xb[(long)ka * HW + n0 + col];
    b.y = Xb[(long)(ka + 1) * HW + n0 + col];
    if (k + 4 < 128)
      __builtin_prefetch(&Xb[(long)(k + 4 + 2 * hi) * HW + n0 + col], 0, 1);
    acc = __builtin_amdgcn_wmma_f32_16x16x4_f32(
        /*neg_a=*/false, a, /*neg_b=*/false, b,
        /*c_mod=*/(short)0, acc, /*reuse_a=*/false, /*reuse_b=*/false);
  }
  // C/D layout: VGPR r -> M = r (lanes 0-15) / M = 8+r (lanes 16-31), N = col
#pragma unroll
  for (int r = 0; r < 8; ++r) {
    Yb[(long)(m0 + 8 * hi + r) * HW + n0 + col] = acc[r];
  }
}

// ---------------------------------------------------------------------------
// Kernel 3: generic 256-point radix-2 FFT pass. 32 lines per 256-thread block,
// LDS-resident. Forward = DIF (natural in, bit-rev out), inverse = DIT
// (bit-rev in, natural out) -> no explicit bit-reversal pass is ever needed.
// Zero-pad beyond n_valid on load; store only n_store outputs.
// Line addressing: off(l) = (l / div) * os + (l % div) * is, element stride es.
// ---------------------------------------------------------------------------
__global__ void fft256_pass_kernel(const float2* __restrict__ cin,
                                   const float* __restrict__ rin,
                                   float2* __restrict__ out,
                                   int in_div, int in_os, int in_is, int in_es,
                                   int out_div, int out_os, int out_is, int out_es,
                                   int n_valid, int n_store,
                                   int fwd, int contig, int real_in) {
  __shared__ float2 lds[NFFT * LDS_STRIDE];
  const int tid = threadIdx.x;
  const int lbase = blockIdx.x * 32;
  const int line = tid & 31;
  const int sub  = tid >> 5;   // 0..7

  // ---- load (coalesced per layout) ----
  if (contig) {
    for (int it = 0; it < 32; ++it) {
      const int idx = it * 256 + tid;
      const int lloc = idx >> 8;
      const int e = idx & 255;
      const int l = lbase + lloc;
      const long off = (long)(l / in_div) * in_os + (long)(l % in_div) * in_is;
      float2 v;
      if (e < n_valid) {
        if (real_in) { v.x = rin[off + e]; v.y = 0.f; }
        else         { v = cin[off + (long)e * in_es]; }
      } else { v.x = 0.f; v.y = 0.f; }
      lds[e * LDS_STRIDE + lloc] = v;
    }
  } else {
    const int l = lbase + line;
    const long off = (long)(l / in_div) * in_os + (long)(l % in_div) * in_is;
    for (int it = 0; it < 32; ++it) {
      const int e = it * 8 + sub;
      float2 v;
      if (e < n_valid) v = cin[off + (long)e * in_es];
      else { v.x = 0.f; v.y = 0.f; }
      lds[e * LDS_STRIDE + line] = v;
    }
  }

  // ---- butterflies: each thread owns one line-lane, 16 butterflies/stage ----
  if (fwd) {
    for (int s = 8; s >= 1; --s) {               // DIF: len = 256 .. 2
      const int half = 1 << (s - 1);
      __syncthreads();
      for (int j = sub; j < 128; j += 8) {
        const int k = j & (half - 1);
        const int i0 = ((j >> (s - 1)) << s) + k;
        const int i1 = i0 + half;
        const float2 u = lds[i0 * LDS_STRIDE + line];
        const float2 v = lds[i1 * LDS_STRIDE + line];
        float2 sum, dif;
        sum.x = u.x + v.x; sum.y = u.y + v.y;
        dif.x = u.x - v.x; dif.y = u.y - v.y;
        const float ang = -6.28318530717958647692f * (float)k / (float)(1 << s);
        float sn, cs;
        __sincosf(ang, &sn, &cs);
        float2 tw;
        tw.x = dif.x * cs - dif.y * sn;
        tw.y = dif.x * sn + dif.y * cs;
        lds[i0 * LDS_STRIDE + line] = sum;
        lds[i1 * LDS_STRIDE + line] = tw;
      }
    }
  } else {
    for (int s = 1; s <= 8; ++s) {               // DIT: len = 2 .. 256
      const int half = 1 << (s - 1);
      __syncthreads();
      for (int j = sub; j < 128; j += 8) {
        const int k = j & (half - 1);
        const int i0 = ((j >> (s - 1)) << s) + k;
        const int i1 = i0 + half;
        const float2 u = lds[i0 * LDS_STRIDE + line];
        const float2 v = lds[i1 * LDS_STRIDE + line];
        const float ang = 6.28318530717958647692f * (float)k / (float)(1 << s);
        float sn, cs;
        __sincosf(ang, &sn, &cs);
        float2 vt, r0, r1;
        vt.x = v.x * cs - v.y * sn;
        vt.y = v.x * sn + v.y * cs;
        r0.x = u.x + vt.x; r0.y = u.y + vt.y;
        r1.x = u.x - vt.x; r1.y = u.y - vt.y;
        lds[i0 * LDS_STRIDE + line] = r0;
        lds[i1 * LDS_STRIDE + line] = r1;
      }
    }
  }
  __syncthreads();

  // ---- store (crop to n_store) ----
  if (contig) {
    for (int it = 0; it < 32; ++it) {
      const int idx = it * 256 + tid;
      const int lloc = idx >> 8;
      const int e = idx & 255;
      if (e < n_store) {
        const int l = lbase + lloc;
        const long off = (long)(l / out_div) * out_os + (long)(l % out_div) * out_is;
        out[off + (long)e * out_es] = lds[e * LDS_STRIDE + lloc];
      }
    }
  } else {
    const int l = lbase + line;
    const long off = (long)(l / out_div) * out_os + (long)(l % out_div) * out_is;
    for (int it = 0; it < 32; ++it) {
      const int e = it * 8 + sub;
      if (e < n_store) out[off + (long)e * out_es] = lds[e * LDS_STRIDE + line];
    }
  }
}

// ---------------------------------------------------------------------------
// Kernel 4: spectral multiply. Forward spectrum is stored bit-reversed in each
// axis (DIF), so index invpsf at bit-reversed coords; fold in 1/256^3.
// invpsf is conj(fft(psf)) already -> plain complex multiply.
// ---------------------------------------------------------------------------
__global__ void spectral_mul_kernel(float2* __restrict__ c3,
                                    const float* __restrict__ pr,
                                    const float* __restrict__ pi) {
  const long i = (long)blockIdx.x * blockDim.x + threadIdx.x;  // 0 .. 256^3-1
  const int w = (int)(i & 255);
  const int h = (int)((i >> 8) & 255);
  const int m = (int)(i >> 16);
  const int rw = (int)(__brev((unsigned)w) >> 24);
  const int rh = (int)(__brev((unsigned)h) >> 24);
  const int rm = (int)(__brev((unsigned)m) >> 24);
  const long ridx = ((long)rm << 16) | ((long)rh << 8) | (long)rw;
  const float a = pr[ridx] * SPEC_SCALE;
  const float b = pi[ridx] * SPEC_SCALE;
#pragma unroll
  for (int bb = 0; bb < 4; ++bb) {
    const long idx = (long)bb * 16777216L + i;
    const float2 v = c3[idx];
    float2 r;
    r.x = v.x * a - v.y * b;
    r.y = v.x * b + v.y * a;
    c3[idx] = r;
  }
}

// ---------------------------------------------------------------------------
// Kernel 5: complex-square magnitude nonlinearity.
// sq = vol[g]^2 + vol[g+2]^2 (complex squares), out = sqrt(relu((|sq|+Re)/2))
// ---------------------------------------------------------------------------
__global__ void mag_kernel(const float2* __restrict__ vol,
                           float* __restrict__ ssum) {
  const long i = (long)blockIdx.x * blockDim.x + threadIdx.x;  // 0 .. 2*128^3-1
  const long vox = 2097152L;                                   // 128^3
  const int g = (int)(i / vox);
  const long p = i - (long)g * vox;
  const float2 v0 = vol[(long)g * vox + p];
  const float2 v1 = vol[(long)(g + 2) * vox + p];
  const float sr = v0.x * v0.x - v0.y * v0.y + v1.x * v1.x - v1.y * v1.y;
  const float si = 2.f * (v0.x * v0.y + v1.x * v1.y);
  const float mag = sqrtf(sr * sr + si * si);
  const float val = 0.5f * (mag + sr);
  ssum[i] = sqrtf(val > 0.f ? val : 0.f);
}

// ---------------------------------------------------------------------------
// Launch: inputs (setup_inputs order):
// 0 feat (1,2,96,128,128) f32 | 1 wave (2,1,31) f32 | 2 mtx (128,128) f32
// 3 mtxi (128,128) f32 | 4 invpsf_real (256^3) | 5 invpsf_imag (256^3)
// 6 tbes i32 | 7 tens i32.   Output: (1,2,128,128,128) f32.
// ---------------------------------------------------------------------------
extern "C" void kernel_launch(void* const* d_in, const int* in_sizes, int n_in,
                              void* d_out, int out_size, void* d_ws, size_t ws_size,
                              hipStream_t stream) {
  const float* feat = (const float*)d_in[0];
  const float* wave = (const float*)d_in[1];
  const float* mtx  = (const float*)d_in[2];
  const float* mtxi = (const float*)d_in[3];
  const float* pr   = (const float*)d_in[4];
  const float* pi   = (const float*)d_in[5];
  const int*   tbes = (const int*)d_in[6];
  (void)in_sizes; (void)n_in; (void)out_size; (void)ws_size;

  char* base = (char*)d_ws;
  // [0, 32MB)  data2 ; [32MB, 64MB) tmp ; later both reused as vol (64MB)
  float*  data2 = (float*)(base);
  float*  tmp   = (float*)(base + 33554432L);
  float2* vol   = (float2*)(base);
  // R1: 512MB region, holds c1 (128MB) then c3 (512MB) then c1' (128MB)
  float2* R1 = (float2*)(base + 67108864L);
  // R2: 256MB region, holds c2 both directions; later reused as sqrt_sum
  float2* R2 = (float2*)(base + 67108864L + 536870912L);
  float*  ssum = (float*)R2;

  const dim3 b256(256);
  const int BIGDIV = 1 << 30;  // contiguous passes: l % BIGDIV == l

  // 1) temporal conv -> data2 (4,128,16384)
  conv_wave_kernel<<<dim3(HW / 256, MDIM, 2), b256, 0, stream>>>(feat, wave, tbes, data2);

  // 2) tmp[b] = mtx * data2[b]   (4 batches, WMMA fp32)
  gemm128_wmma_kernel<<<dim3(HW / 64, MDIM / 16, 4), dim3(128), 0, stream>>>(mtx, data2, tmp);

  // 3) forward FFT W: tmp real (4,128,128,128) -> c1 (4,128,128,256)
  fft256_pass_kernel<<<dim3(65536 / 32), b256, 0, stream>>>(
      nullptr, tmp, R1,
      BIGDIV, 0, 128, 1,   BIGDIV, 0, 256, 1,
      128, 256, /*fwd=*/1, /*contig=*/1, /*real_in=*/1);

  // 4) forward FFT H: c1 -> c2 (4,128,256,256)
  fft256_pass_kernel<<<dim3(131072 / 32), b256, 0, stream>>>(
      R1, nullptr, R2,
      256, 32768, 1, 256,   256, 65536, 1, 256,
      128, 256, 1, 0, 0);

  // 5) forward FFT M: c2 -> c3 (4,256,256,256)
  fft256_pass_kernel<<<dim3(262144 / 32), b256, 0, stream>>>(
      R2, nullptr, R1,
      65536, 8388608, 1, 65536,   65536, 16777216, 1, 65536,
      128, 256, 1, 0, 0);

  // 6) spectral multiply (bit-reversed invpsf index, 1/N^3 scale), 4 batches
  spectral_mul_kernel<<<dim3(16777216 / 256), b256, 0, stream>>>(R1, pr, pi);

  // 7) inverse FFT M: c3 -> c2' (4,128,256,256)   (crop to M=128)
  fft256_pass_kernel<<<dim3(262144 / 32), b256, 0, stream>>>(
      R1, nullptr, R2,
      65536, 16777216, 1, 65536,   65536, 8388608, 1, 65536,
      256, 128, 0, 0, 0);

  // 8) inverse FFT H: c2' -> c1' (4,128,128,256)  (crop to H=128)
  fft256_pass_kernel<<<dim3(131072 / 32), b256, 0, stream>>>(
      R2, nullptr, R1,
      256, 65536, 1, 256,   256, 32768, 1, 256,
      256, 128, 0, 0, 0);

  // 9) inverse FFT W: c1' -> vol (4,128,128,128) complex (crop to W=128)
  fft256_pass_kernel<<<dim3(65536 / 32), b256, 0, stream>>>(
      R1, nullptr, vol,
      BIGDIV, 0, 256, 1,   BIGDIV, 0, 128, 1,
      256, 128, 0, 1, 0);

  // 10) magnitude nonlinearity -> sqrt_sum (2,128,16384)
  mag_kernel<<<dim3(4194304 / 256), b256, 0, stream>>>(vol, ssum);

  // 11) out[bd] = mtxi * sqrt_sum[bd]  (2 batches, WMMA fp32)
  gemm128_wmma_kernel<<<dim3(HW / 64, MDIM / 16, 2), dim3(128), 0, stream>>>(
      mtxi, ssum, (float*)d_out);
}